// GNN_18897856102536
// MI455X (gfx1250) — compile-verified
//
#include <hip/hip_runtime.h>
#include <hip/hip_bf16.h>
#include <float.h>

typedef __attribute__((ext_vector_type(2))) float v2f;
typedef __attribute__((ext_vector_type(8))) float v8f;

constexpr int NN   = 40000;   // nodes
constexpr int EE   = 640000;  // edges
constexpr int GG   = 64;      // graphs
constexpr int HD   = 128;     // hidden dim (== DIN)
constexpr int DOUT = 64;
constexpr float LN_EPS = 1e-5f;

// ---------------- wave32 reductions ----------------
__device__ __forceinline__ float wave_sum(float x) {
#pragma unroll
  for (int off = 16; off > 0; off >>= 1) x += __shfl_xor(x, off, 32);
  return x;
}
__device__ __forceinline__ float wave_max(float x) {
#pragma unroll
  for (int off = 16; off > 0; off >>= 1) x = fmaxf(x, __shfl_xor(x, off, 32));
  return x;
}

// ---------------- degree / norm ----------------
__global__ void k_init_dinv(float* dinv) {
  int i = blockIdx.x * blockDim.x + threadIdx.x;
  if (i < NN) dinv[i] = 1.0f;            // self-loop contributes 1 to degree
}
__global__ void k_deg_acc(const int* __restrict__ dst, float* __restrict__ dinv) {
  int e = blockIdx.x * blockDim.x + threadIdx.x;
  if (e < EE) atomicAdd(&dinv[dst[e]], 1.0f);
}
__global__ void k_fin_dinv(float* dinv) {
  int i = blockIdx.x * blockDim.x + threadIdx.x;
  if (i < NN) dinv[i] = rsqrtf(dinv[i]);   // deg >= 1 always (self loop)
}

// ---------------- weight pre-pack ----------------
// Wt[k/2][n][2] = { W[k][n], W[k+1][n] }  -> B fragment becomes one b64 load
template <int NCOLS>
__global__ void k_pack_w(const float* __restrict__ W, float* __restrict__ Wt) {
  int idx = blockIdx.x * blockDim.x + threadIdx.x;       // (K/2)*NCOLS
  if (idx >= 64 * NCOLS) return;
  int kp = idx / NCOLS, n = idx % NCOLS;
  Wt[(size_t)kp * NCOLS * 2 + n * 2 + 0] = W[(size_t)(2 * kp) * NCOLS + n];
  Wt[(size_t)kp * NCOLS * 2 + n * 2 + 1] = W[(size_t)(2 * kp + 1) * NCOLS + n];
}

// ---------------- fp32 WMMA GEMM: C[M x NCOLS] = A[M x 128] @ W[128 x NCOLS] (+bias) ----
// 8 waves/block; each wave computes a 16x32 strip (two 16x16 tiles) so every
// A fragment feeds two v_wmma_f32_16x16x4_f32 on independent accumulators.
// A strip staged in LDS (float4, coalesced); B from pair-packed Wt (b64 loads).
template <int NCOLS, bool BIAS>
__global__ __launch_bounds__(256) void k_gemm_wmma(
    const float* __restrict__ A, const float* __restrict__ Wt,
    const float* __restrict__ bias, float* __restrict__ C) {
  constexpr int K = 128;
  constexpr int CGRP = NCOLS / 32;          // 32-col groups: 4 (H=128) or 2 (64)
  constexpr int RTILES = 8 / CGRP;          // row tiles per block: 2 or 4
  constexpr int ROWS = RTILES * 16;         // 32 or 64 rows per block
  __shared__ float As[ROWS * K];

  const int tid  = threadIdx.x;
  const int wave = tid >> 5;
  const int lane = tid & 31;
  const int half = lane >> 4;               // 0: lanes 0-15, 1: lanes 16-31
  const int l16  = lane & 15;
  const int block_row0 = blockIdx.x * ROWS;

  // cooperative float4 stage of the A strip
#pragma unroll
  for (int i = tid; i < ROWS * 32; i += 256) {
    ((float4*)As)[i] = ((const float4*)A)[(size_t)block_row0 * 32 + i];
  }
  __syncthreads();

  const int trow = wave / CGRP;             // local 16-row tile
  const int n0 = (wave % CGRP) * 32;        // 32-wide col group
  const float* Arow = &As[(trow * 16 + l16) * K];
  const int koff = half * 2;

  v8f acc0 = {}, acc1 = {};
#pragma unroll
  for (int k = 0; k < K; k += 4) {
    v2f a;
    a.x = Arow[k + koff];
    a.y = Arow[k + koff + 1];
    const float* wp = Wt + (size_t)((k + koff) >> 1) * (NCOLS * 2);
    v2f b0 = *(const v2f*)(wp + (n0 + l16) * 2);
    v2f b1 = *(const v2f*)(wp + (n0 + 16 + l16) * 2);
    acc0 = __builtin_amdgcn_wmma_f32_16x16x4_f32(false, a, false, b0,
                                                 (short)0, acc0, false, false);
    acc1 = __builtin_amdgcn_wmma_f32_16x16x4_f32(false, a, false, b1,
                                                 (short)0, acc1, false, false);
  }

  const int gm0 = block_row0 + trow * 16;
  const int c0 = n0 + l16, c1 = n0 + 16 + l16;
  const float bv0 = BIAS ? bias[c0] : 0.0f;
  const float bv1 = BIAS ? bias[c1] : 0.0f;
#pragma unroll
  for (int v = 0; v < 8; ++v) {
    int row = gm0 + v + half * 8;           // C/D layout: M = v + 8*(lane/16)
    C[(size_t)row * NCOLS + c0] = acc0[v] + bv0;
    C[(size_t)row * NCOLS + c1] = acc1[v] + bv1;
  }
}

// ---------------- propagate ----------------
// self-loop term: agg[n] = P[n] * dinv[n]^2   (full overwrite -> no memset needed)
__global__ void k_init_agg(const float* __restrict__ P, const float* __restrict__ dinv,
                           float* __restrict__ agg) {
  int idx = blockIdx.x * blockDim.x + threadIdx.x;   // over NN*32 float4 groups
  if (idx >= NN * 32) return;
  int n = idx >> 5;
  float s = dinv[n]; s = s * s;
  float4 v = ((const float4*)P)[idx];
  ((float4*)agg)[idx] = make_float4(v.x * s, v.y * s, v.z * s, v.w * s);
}
// edges: agg[dst] += P[src] * dinv[src]*dinv[dst]; 32 lanes x float4 per edge
__global__ void k_scatter(const float* __restrict__ P, const float* __restrict__ dinv,
                          const int* __restrict__ src, const int* __restrict__ dst,
                          float* __restrict__ agg) {
  int idx = blockIdx.x * blockDim.x + threadIdx.x;
  if (idx >= EE * 32) return;
  int e = idx >> 5, lane = idx & 31;
  int s = src[e], d = dst[e];
  float nrm = dinv[s] * dinv[d];
  float4 v = ((const float4*)P)[(size_t)s * 32 + lane];
  float* o = agg + (size_t)d * HD + lane * 4;
  atomicAdd(o + 0, v.x * nrm);
  atomicAdd(o + 1, v.y * nrm);
  atomicAdd(o + 2, v.z * nrm);
  atomicAdd(o + 3, v.w * nrm);
}

// ---------------- epilogues ----------------
__device__ __forceinline__ void ln128(float4& v, const float4 g, const float4 b) {
  float mu = wave_sum(v.x + v.y + v.z + v.w) * (1.0f / 128.0f);
  float dx = v.x - mu, dy = v.y - mu, dz = v.z - mu, dw = v.w - mu;
  float var = wave_sum(dx * dx + dy * dy + dz * dz + dw * dw) * (1.0f / 128.0f);
  float r = rsqrtf(var + LN_EPS);
  v.x = dx * r * g.x + b.x;
  v.y = dy * r * g.y + b.y;
  v.z = dz * r * g.z + b.z;
  v.w = dw * r * g.w + b.w;
}

// h = LN2(LN1(relu(agg + b1))) ; one wave32 per 128-wide row
__global__ void k_epilogue1(const float* __restrict__ agg, const float* __restrict__ bias,
                            const float* __restrict__ g1, const float* __restrict__ be1,
                            const float* __restrict__ g2, const float* __restrict__ be2,
                            float* __restrict__ out) {
  int wave = threadIdx.x >> 5, lane = threadIdx.x & 31;
  int row = blockIdx.x * 8 + wave;
  if (row >= NN) return;
  float4 v = ((const float4*)(agg + (size_t)row * HD))[lane];
  float4 bb = ((const float4*)bias)[lane];
  v.x = fmaxf(v.x + bb.x, 0.0f);
  v.y = fmaxf(v.y + bb.y, 0.0f);
  v.z = fmaxf(v.z + bb.z, 0.0f);
  v.w = fmaxf(v.w + bb.w, 0.0f);
  ln128(v, ((const float4*)g1)[lane], ((const float4*)be1)[lane]);
  ln128(v, ((const float4*)g2)[lane], ((const float4*)be2)[lane]);
  ((float4*)(out + (size_t)row * HD))[lane] = v;
}

__global__ void k_bias_add(const float* __restrict__ agg, const float* __restrict__ bias,
                           float* __restrict__ out) {
  int idx = blockIdx.x * blockDim.x + threadIdx.x;   // NN*32 float4 groups
  if (idx >= NN * 32) return;
  int lane32 = idx & 31;
  float4 v = ((const float4*)agg)[idx];
  float4 b = ((const float4*)bias)[lane32];
  ((float4*)out)[idx] = make_float4(v.x + b.x, v.y + b.y, v.z + b.z, v.w + b.w);
}

// ---------------- pooling + log_softmax ----------------
__global__ void k_init_pool(float* gmax, float* gsum, float* cnt) {
  int i = blockIdx.x * blockDim.x + threadIdx.x;
  if (i < GG * DOUT) { gmax[i] = -FLT_MAX; gsum[i] = 0.0f; }
  if (i < GG) cnt[i] = 0.0f;
}
__global__ void k_pool(const float* __restrict__ node, const int* __restrict__ batch,
                       float* __restrict__ gmax, float* __restrict__ gsum,
                       float* __restrict__ cnt) {
  int idx = blockIdx.x * blockDim.x + threadIdx.x;   // NN*DOUT
  if (idx >= NN * DOUT) return;
  int n = idx >> 6, f = idx & 63;
  int b = batch[n];
  float v = node[idx];
  atomicMax(&gmax[b * DOUT + f], v);
  atomicAdd(&gsum[b * DOUT + f], v);
  if (f == 0) atomicAdd(&cnt[b], 1.0f);
}
__global__ void k_final_lsm(const float* __restrict__ gmax, const float* __restrict__ gsum,
                            const float* __restrict__ cnt, float* __restrict__ out) {
  int wave = threadIdx.x >> 5, lane = threadIdx.x & 31;
  int g = blockIdx.x * 8 + wave;
  if (g >= GG) return;
  float c = fmaxf(cnt[g], 1.0f);
  float4 v;
  if (lane < 16) {
    v = ((const float4*)(gmax + g * DOUT))[lane];
  } else {
    float4 s = ((const float4*)(gsum + g * DOUT))[lane - 16];
    float ic = 1.0f / c;
    v = make_float4(s.x * ic, s.y * ic, s.z * ic, s.w * ic);
  }
  float m = wave_max(fmaxf(fmaxf(v.x, v.y), fmaxf(v.z, v.w)));
  float e = wave_sum(expf(v.x - m) + expf(v.y - m) + expf(v.z - m) + expf(v.w - m));
  float lse = m + logf(e);
  ((float4*)(out + g * 2 * DOUT))[lane] =
      make_float4(v.x - lse, v.y - lse, v.z - lse, v.w - lse);
}

// ---------------- launcher ----------------
extern "C" void kernel_launch(void* const* d_in, const int* in_sizes, int n_in,
                              void* d_out, int out_size, void* d_ws, size_t ws_size,
                              hipStream_t stream) {
  const float* x     = (const float*)d_in[0];
  const int*   ei    = (const int*)d_in[1];
  const int*   src   = ei;
  const int*   dst   = ei + EE;
  const int*   batch = (const int*)d_in[2];
  const float* W1 = (const float*)d_in[3],  *b1  = (const float*)d_in[4];
  const float* g1 = (const float*)d_in[5],  *be1 = (const float*)d_in[6];
  const float* g2 = (const float*)d_in[7],  *be2 = (const float*)d_in[8];
  const float* W2 = (const float*)d_in[9],  *b2  = (const float*)d_in[10];
  const float* W3 = (const float*)d_in[11], *b3  = (const float*)d_in[12];
  const float* Wp1 = (const float*)d_in[13], *bp1 = (const float*)d_in[14];
  const float* Wp2 = (const float*)d_in[15], *bp2 = (const float*)d_in[16];
  float* out = (float*)d_out;

  // workspace layout (floats):
  // dinv | P | AGG | H | NODE | GMAX | GSUM | CNT | W1t W2t W3t Wp1t | Wp2t
  float* dinv = (float*)d_ws;
  float* P    = dinv + NN;
  float* AGG  = P + (size_t)NN * HD;
  float* Hb   = AGG + (size_t)NN * HD;
  float* NODE = Hb + (size_t)NN * HD;
  float* GMAX = NODE + (size_t)NN * DOUT;
  float* GSUM = GMAX + GG * DOUT;
  float* CNT  = GSUM + GG * DOUT;
  float* W1t  = CNT + GG;
  float* W2t  = W1t + HD * HD;
  float* W3t  = W2t + HD * HD;
  float* Wp1t = W3t + HD * HD;
  float* Wp2t = Wp1t + HD * HD;

  const int nb_nodes = (NN + 255) / 256;          // 157
  const int nb_e32   = (EE * 32) / 256;           // 80000
  const int nb_nv4   = (NN * 32) / 256;           // 5000
  const int nb_rows  = NN / 8;                    // 5000 (LN rows, 8 waves/blk)

  // pair-pack weights for b64 B-fragment loads
  k_pack_w<HD><<<(64 * HD + 255) / 256, 256, 0, stream>>>(W1, W1t);
  k_pack_w<HD><<<(64 * HD + 255) / 256, 256, 0, stream>>>(W2, W2t);
  k_pack_w<HD><<<(64 * HD + 255) / 256, 256, 0, stream>>>(W3, W3t);
  k_pack_w<HD><<<(64 * HD + 255) / 256, 256, 0, stream>>>(Wp1, Wp1t);
  k_pack_w<DOUT><<<(64 * DOUT + 255) / 256, 256, 0, stream>>>(Wp2, Wp2t);

  // normalization coefficients
  k_init_dinv<<<nb_nodes, 256, 0, stream>>>(dinv);
  k_deg_acc<<<(EE + 255) / 256, 256, 0, stream>>>(dst, dinv);
  k_fin_dinv<<<nb_nodes, 256, 0, stream>>>(dinv);

  // GCN layer 1
  k_gemm_wmma<HD, false><<<NN / 32, 256, 0, stream>>>(x, W1t, nullptr, P);
  k_init_agg<<<nb_nv4, 256, 0, stream>>>(P, dinv, AGG);
  k_scatter<<<nb_e32, 256, 0, stream>>>(P, dinv, src, dst, AGG);
  k_epilogue1<<<nb_rows, 256, 0, stream>>>(AGG, b1, g1, be1, g2, be2, Hb);

  // GCN layer 2
  k_gemm_wmma<HD, false><<<NN / 32, 256, 0, stream>>>(Hb, W2t, nullptr, P);
  k_init_agg<<<nb_nv4, 256, 0, stream>>>(P, dinv, AGG);
  k_scatter<<<nb_e32, 256, 0, stream>>>(P, dinv, src, dst, AGG);
  k_bias_add<<<nb_nv4, 256, 0, stream>>>(AGG, b2, Hb);

  // GCN layer 3
  k_gemm_wmma<HD, false><<<NN / 32, 256, 0, stream>>>(Hb, W3t, nullptr, P);
  k_init_agg<<<nb_nv4, 256, 0, stream>>>(P, dinv, AGG);
  k_scatter<<<nb_e32, 256, 0, stream>>>(P, dinv, src, dst, AGG);
  k_bias_add<<<nb_nv4, 256, 0, stream>>>(AGG, b3, Hb);

  // MLP head
  k_gemm_wmma<HD, true><<<NN / 32, 256, 0, stream>>>(Hb, Wp1t, bp1, P);
  k_gemm_wmma<DOUT, true><<<NN / 64, 256, 0, stream>>>(P, Wp2t, bp2, NODE);

  // pooling + log_softmax
  k_init_pool<<<(GG * DOUT + 255) / 256, 256, 0, stream>>>(GMAX, GSUM, CNT);
  k_pool<<<(NN * DOUT) / 256, 256, 0, stream>>>(NODE, batch, GMAX, GSUM, CNT);
  k_final_lsm<<<GG / 8, 256, 0, stream>>>(GMAX, GSUM, CNT, out);
}